// GAFM_15023795602158
// MI455X (gfx1250) — compile-verified
//
#include <hip/hip_runtime.h>
#include <math.h>

// ---------------------------------------------------------------------------
// GAFM 2-hop graph FM, MI455X (gfx1250, wave32).
// Attention path is mathematically the identity (softmax over a singleton
// axis == 1.0), so Wa/ba/Wh/bh are dead inputs. The live work is a huge
// random-row gather -> renorm -> FM reduce, i.e. bandwidth-bound. We use the
// CDNA5 async global->LDS gather path (global_load_async_to_lds_b128 +
// s_wait_asynccnt) to stage each node's 32x64 f32 neighbor tile in LDS.
// ---------------------------------------------------------------------------

#define WPB 8  // waves per block (256 threads)

__device__ __forceinline__ float wave_sum32(float v) {
#pragma unroll
  for (int m = 16; m > 0; m >>= 1) v += __shfl_xor(v, m, 32);
  return v;
}

__device__ __forceinline__ float bcast_lane(float v, int srclane) {
  return __uint_as_float(__builtin_amdgcn_readlane(__float_as_uint(v), srclane));
}

__device__ __forceinline__ float renorm_scale(float n2) {
  float n = sqrtf(n2);
  return fminf(1.0f, 1.0f / fmaxf(n, 1e-12f));
}

// ---- hop 0: agg0[node] = fm(renorm(entity[nbr0[node,:]])) + renorm(entity[node_ids0[node]])
__global__ __launch_bounds__(256) void gafm_hop0(
    const float* __restrict__ entity,     // [N_ENT, 64]
    const int* __restrict__ node_ids0,    // [N0]
    const int* __restrict__ nbr0,         // [N0, 32]
    float* __restrict__ agg0,             // [N0, 64]  (workspace)
    int n0) {
  __shared__ float lds[WPB * 32 * 64];    // 64 KB: per wave 32 rows x 64 f32

  const int lane = threadIdx.x & 31;
  const int wave = threadIdx.x >> 5;
  const int node = blockIdx.x * WPB + wave;  // uniform per wave
  if (node >= n0) return;

  float* wtile = &lds[wave * 2048];
  const unsigned ldsBase = (unsigned)(uintptr_t)wtile;  // low 32 bits = LDS byte addr
  const unsigned long long ebase = (unsigned long long)(uintptr_t)entity;

  // Per-lane neighbor index (coalesced), then async-gather 32 rows into LDS.
  // Each async b128 instruction fetches TWO complete rows fully coalesced:
  // lanes 0..15 cover row 2c, lanes 16..31 cover row 2c+1 (16B chunks each).
  const int idx = nbr0[(size_t)node * 32 + lane];
  const unsigned chunkOff = (unsigned)(lane & 15) << 4;  // 0..240
  const int hi = lane >> 4;

#pragma unroll
  for (int c = 0; c < 16; ++c) {
    int r0 = __builtin_amdgcn_readlane(idx, 2 * c);
    int r1 = __builtin_amdgcn_readlane(idx, 2 * c + 1);
    unsigned row = (unsigned)(hi ? r1 : r0);
    unsigned long long g = ebase + (((unsigned long long)row) << 8) + chunkOff;
    unsigned l = ldsBase + (unsigned)(c << 9) + ((unsigned)lane << 4);
    asm volatile("global_load_async_to_lds_b128 %0, %1, off"
                 :: "v"(l), "v"(g) : "memory");
  }

  // Overlap: target row lookup + renorm while the gather is in flight.
  const int tnode = node_ids0[node];
  float2 t = ((const float2*)(entity + (size_t)tnode * 64))[lane];
  float tsc = renorm_scale(wave_sum32(t.x * t.x + t.y * t.y));
  t.x *= tsc; t.y *= tsc;

  asm volatile("s_wait_asynccnt 0x0" ::: "memory");

  // Pass A: all 32 row norms in parallel (lane l -> row l), rotated reads
  // so banks 2*((l+c)&31) are conflict-free.
  float n2 = 0.0f;
#pragma unroll
  for (int c = 0; c < 32; ++c) {
    int d = (lane + c) & 31;
    float2 e = *(const float2*)&wtile[lane * 64 + d * 2];
    n2 += e.x * e.x + e.y * e.y;
  }
  const float myscale = renorm_scale(n2);

  // Pass B: scale-weighted FM accumulation. All lanes read row j (banks
  // 2l,2l+1: conflict-free); per-row scale broadcast via v_readlane.
  float2 s = make_float2(0.f, 0.f), ss = make_float2(0.f, 0.f);
#pragma unroll
  for (int j = 0; j < 32; ++j) {
    float sc = bcast_lane(myscale, j);
    float2 e = *(const float2*)&wtile[j * 64 + lane * 2];
    float ex = e.x * sc, ey = e.y * sc;
    s.x += ex; s.y += ey;
    ss.x = fmaf(ex, ex, ss.x); ss.y = fmaf(ey, ey, ss.y);
  }

  float2 o;
  o.x = s.x * s.x - ss.x + t.x;
  o.y = s.y * s.y - ss.y + t.y;
  ((float2*)(agg0 + (size_t)node * 64))[lane] = o;
}

// ---- hop 1: logit[b] = sigmoid( renorm(user[u[b]]) . (fm(agg0[nbr1[b,:]]) + renorm(entity[item[b]])) )
__global__ __launch_bounds__(256) void gafm_hop1(
    const float* __restrict__ entity,
    const float* __restrict__ user_emb,
    const float* __restrict__ agg0,       // [N0, 64]
    const int* __restrict__ u_ids,        // [B]
    const int* __restrict__ item_ids,     // [B]
    const int* __restrict__ nbr1,         // [B, 32]
    float* __restrict__ out,              // [B]
    int nb) {
  const int lane = threadIdx.x & 31;
  const int wave = threadIdx.x >> 5;
  const int b = blockIdx.x * WPB + wave;
  if (b >= nb) return;

  const int idx = nbr1[(size_t)b * 32 + lane];
  float2 s = make_float2(0.f, 0.f), ss = make_float2(0.f, 0.f);
#pragma unroll
  for (int j = 0; j < 32; ++j) {
    int r = __builtin_amdgcn_readlane(idx, j);          // uniform row id
    float2 e = ((const float2*)(agg0 + (size_t)r * 64))[lane];  // coalesced 256B
    s.x += e.x; s.y += e.y;
    ss.x = fmaf(e.x, e.x, ss.x); ss.y = fmaf(e.y, e.y, ss.y);
  }

  const int it = item_ids[b];
  float2 t = ((const float2*)(entity + (size_t)it * 64))[lane];
  float tsc = renorm_scale(wave_sum32(t.x * t.x + t.y * t.y));
  t.x *= tsc; t.y *= tsc;

  float2 items;
  items.x = s.x * s.x - ss.x + t.x;
  items.y = s.y * s.y - ss.y + t.y;

  const int uu = u_ids[b];
  float2 ue = ((const float2*)(user_emb + (size_t)uu * 64))[lane];
  float usc = renorm_scale(wave_sum32(ue.x * ue.x + ue.y * ue.y));
  ue.x *= usc; ue.y *= usc;

  float d = wave_sum32(ue.x * items.x + ue.y * items.y);
  if (lane == 0) out[b] = 1.0f / (1.0f + expf(-d));
}

extern "C" void kernel_launch(void* const* d_in, const int* in_sizes, int n_in,
                              void* d_out, int out_size, void* d_ws, size_t ws_size,
                              hipStream_t stream) {
  // setup_inputs() order:
  // 0 entity_emb f32 [1e6,64]   1 user_emb f32 [1e5,64]
  // 2 Wa 3 ba 4 Wh 5 bh        (dead: softmax over singleton axis == 1.0)
  // 6 u i32 [B]  7 item_ids i32 [B]  8 nbr1_idx i32 [B,32]
  // 9 node_ids0 i32 [N0]  10 nbr0 i32 [N0,32]
  const float* entity   = (const float*)d_in[0];
  const float* user_emb = (const float*)d_in[1];
  const int* u_ids      = (const int*)d_in[6];
  const int* item_ids   = (const int*)d_in[7];
  const int* nbr1       = (const int*)d_in[8];
  const int* node_ids0  = (const int*)d_in[9];
  const int* nbr0       = (const int*)d_in[10];
  float* out = (float*)d_out;

  const int B  = in_sizes[6];   // 2048
  const int N0 = in_sizes[9];   // 65536
  float* agg0 = (float*)d_ws;   // needs N0*64*4 = 16 MB of workspace

  dim3 blk(256);
  dim3 g0((N0 + WPB - 1) / WPB);
  dim3 g1((B + WPB - 1) / WPB);
  gafm_hop0<<<g0, blk, 0, stream>>>(entity, node_ids0, nbr0, agg0, N0);
  gafm_hop1<<<g1, blk, 0, stream>>>(entity, user_emb, agg0, u_ids, item_ids, nbr1, out, B);
}